// RVRelu_841813590061
// MI455X (gfx1250) — compile-verified
//
#include <hip/hip_runtime.h>

// 4 x f32, backed by a clang ext_vector so __builtin_nontemporal_* accepts it
typedef float f32x4 __attribute__((ext_vector_type(4)));

// ---------------------------------------------------------------------------
// mu_out = relu(mu_in)   (tiny: B*D = 131072 floats)
// ---------------------------------------------------------------------------
__global__ void relu_mu_kernel(const float* __restrict__ mu_in,
                               float* __restrict__ mu_out, int n) {
    int idx = blockIdx.x * blockDim.x + threadIdx.x;
    if (idx < n) {
        float v = mu_in[idx];
        mu_out[idx] = v > 0.f ? v : 0.f;
    }
}

// ---------------------------------------------------------------------------
// Sigma_out[b,i,j] = Sigma_in[b,i,j] * 1[mu[b,i]>0] * 1[mu[b,j]>0]
//
// One 256-thread block (8 wave32) per (b,i) row of the DxD matrix, D = 1024.
// Lane t owns float4 columns [4t, 4t+3]  ->  256 lanes * 16 B = 4 KB row.
//
// Row gate mu[b,i] is workgroup-uniform: pull it into an SGPR with
// readfirstlane so the dead-row path is a scalar branch that skips the
// 4 KB Sigma read entirely (~50% of rows) and only streams zeros out.
// Sigma traffic is single-use -> non-temporal (TH=NT) loads/stores keep the
// reused 4 KB mu rows hot in WGP$/L2 instead of being evicted by the stream.
// ---------------------------------------------------------------------------
__global__ void sigma_mask_kernel(const float* __restrict__ mu_in,
                                  const f32x4* __restrict__ Sigma_in,
                                  f32x4* __restrict__ Sigma_out) {
    const int row = blockIdx.x;          // row = b*D + i, 0 .. B*D-1
    const int b   = row >> 10;           // D = 1024
    const int tid = threadIdx.x;         // 0 .. 255
    const size_t rbase = (size_t)row * 256 + tid;   // float4 index

    // Workgroup-uniform row gate -> SGPR -> scalar branch.
    float rowv = mu_in[row];
    rowv = __uint_as_float(__builtin_amdgcn_readfirstlane(__float_as_uint(rowv)));

    if (rowv > 0.f) {
        // Column mask from the (cache-resident, regular-temporal) mu row.
        const f32x4* mu4 = (const f32x4*)(mu_in + ((size_t)b << 10));
        f32x4 m = mu4[tid];
        // Streamed, single-use Sigma row: non-temporal 128-bit load.
        f32x4 s = __builtin_nontemporal_load(&Sigma_in[rbase]);
        f32x4 r;
        r.x = (m.x > 0.f) ? s.x : 0.f;
        r.y = (m.y > 0.f) ? s.y : 0.f;
        r.z = (m.z > 0.f) ? s.z : 0.f;
        r.w = (m.w > 0.f) ? s.w : 0.f;
        __builtin_nontemporal_store(r, &Sigma_out[rbase]);
    } else {
        // Dead row: never touch Sigma_in, just stream zeros out.
        f32x4 z = {0.f, 0.f, 0.f, 0.f};
        __builtin_nontemporal_store(z, &Sigma_out[rbase]);
    }
}

extern "C" void kernel_launch(void* const* d_in, const int* in_sizes, int n_in,
                              void* d_out, int out_size, void* d_ws, size_t ws_size,
                              hipStream_t stream) {
    const float* mu_in    = (const float*)d_in[0];   // (B, D)    f32
    const float* Sigma_in = (const float*)d_in[1];   // (B, D, D) f32

    const int BD = in_sizes[0];                      // B*D = 131072
    const int D  = in_sizes[1] / in_sizes[0];        // 1024

    float* mu_out    = (float*)d_out;                // first output, BD floats
    float* Sigma_out = (float*)d_out + BD;           // second output, BD*D floats

    // mu relu: 512 blocks x 256
    relu_mu_kernel<<<(BD + 255) / 256, 256, 0, stream>>>(mu_in, mu_out, BD);

    // Sigma mask: one block per (b,i) row, D/4 lanes each (256 for D=1024)
    sigma_mask_kernel<<<BD, D / 4, 0, stream>>>(
        mu_in, (const f32x4*)Sigma_in, (f32x4*)Sigma_out);
}